// NATSearch_39049842655290
// MI455X (gfx1250) — compile-verified
//
#include <hip/hip_runtime.h>

typedef __attribute__((ext_vector_type(2))) float v2f;
typedef __attribute__((ext_vector_type(8))) float v8f;

#define NFR   4              // B*T frames
#define HH    64
#define WW    64
#define CC    64
#define C3    192
#define HEADS 2
#define DH    32
#define KW    7
#define PIX   (HH*WW)        // 4096
#define TOK   (NFR*PIX)      // 16384

#define WMMA_F32(a, b, c) \
  __builtin_amdgcn_wmma_f32_16x16x4_f32(false, (a), false, (b), (short)0, (c), false, false)

// ---------------------------------------------------------------------------
// Kernel 1: QKV projection  (TOK x 64) @ (64 x 192) + b_qkv
// One wave per 64x16 output block = 4 row sub-tiles sharing one set of
// preloaded B (weight) fragments. K looped in steps of 4 via WMMA f32
// 16x16x4 -> 64 v_wmma per wave, 2 global loads per wmma in steady state.
// vid layout (n, c, y, x): channel stride = PIX, so A-fragment loads are
// coalesced across lanes (consecutive tokens along x).
// ---------------------------------------------------------------------------
__global__ __launch_bounds__(256) void qkv_gemm_kernel(
    const float* __restrict__ vid, const float* __restrict__ w_qkv,
    const float* __restrict__ b_qkv, float* __restrict__ qkv)
{
  const int wave = (blockIdx.x * blockDim.x + threadIdx.x) >> 5;
  const int lane = threadIdx.x & 31;
  const int half = lane >> 4;       // 0: K={k0,k0+1}, 1: K={k0+2,k0+3}
  const int lr   = lane & 15;       // M row (A) / N col (B, D)

  const int row_blk  = wave / (C3 / 16);   // 64-row block index
  const int col_tile = wave % (C3 / 16);
  const int row0 = row_blk * 64;
  const int col0 = col_tile * 16;

  __builtin_prefetch(&w_qkv[col0], 0, 0);  // global_prefetch_b8 on the weight tile

  // Preload all 16 B fragments once; reused by the 4 row sub-tiles.
  v2f bf[16];
#pragma unroll
  for (int s = 0; s < 16; ++s) {
    const int k0 = 4 * s + 2 * half;
    bf[s].x = w_qkv[(k0    ) * C3 + col0 + lr];
    bf[s].y = w_qkv[(k0 + 1) * C3 + col0 + lr];
  }

  // 64 rows stay inside one frame (PIX % 64 == 0).
  const int frm = row0 / PIX;
  const float* f0 = vid + (size_t)frm * CC * PIX;      // + c*PIX walks channels
  const float* aB0 = f0 + ((row0 +  0 + lr) % PIX);
  const float* aB1 = f0 + ((row0 + 16 + lr) % PIX);
  const float* aB2 = f0 + ((row0 + 32 + lr) % PIX);
  const float* aB3 = f0 + ((row0 + 48 + lr) % PIX);

  v8f acc0 = {}, acc1 = {}, acc2 = {}, acc3 = {};
#pragma unroll
  for (int s = 0; s < 16; ++s) {
    const int k0 = 4 * s + 2 * half;
    v2f a0, a1, a2, a3;
    a0.x = aB0[(size_t)(k0) * PIX]; a0.y = aB0[(size_t)(k0 + 1) * PIX];
    a1.x = aB1[(size_t)(k0) * PIX]; a1.y = aB1[(size_t)(k0 + 1) * PIX];
    a2.x = aB2[(size_t)(k0) * PIX]; a2.y = aB2[(size_t)(k0 + 1) * PIX];
    a3.x = aB3[(size_t)(k0) * PIX]; a3.y = aB3[(size_t)(k0 + 1) * PIX];
    acc0 = WMMA_F32(a0, bf[s], acc0);
    acc1 = WMMA_F32(a1, bf[s], acc1);
    acc2 = WMMA_F32(a2, bf[s], acc2);
    acc3 = WMMA_F32(a3, bf[s], acc3);
  }

  const float bias = b_qkv[col0 + lr];
#pragma unroll
  for (int r = 0; r < 8; ++r) {
    const int m = r + 8 * half;     // C/D layout: VGPR r -> M = r (+8 for hi half)
    qkv[(size_t)(row0 +  0 + m) * C3 + col0 + lr] = acc0[r] + bias;
    qkv[(size_t)(row0 + 16 + m) * C3 + col0 + lr] = acc1[r] + bias;
    qkv[(size_t)(row0 + 32 + m) * C3 + col0 + lr] = acc2[r] + bias;
    qkv[(size_t)(row0 + 48 + m) * C3 + col0 + lr] = acc3[r] + bias;
  }
}

// ---------------------------------------------------------------------------
// Kernel 2: neighborhood attention with online softmax.
// One thread per (frame, head, pixel); clamped 7x7 window, dh=32.
// Output stored channel-major (attn_t[c][token]) so the projection GEMM's
// A-fragment loads are coalesced.
// ---------------------------------------------------------------------------
__global__ __launch_bounds__(256) void natten_kernel(
    const float* __restrict__ qkv, const float* __restrict__ rpb,
    float* __restrict__ attn_t)
{
  const int tid = blockIdx.x * blockDim.x + threadIdx.x;   // (frm*HEADS+h)*PIX + pix
  const int pix = tid & (PIX - 1);
  const int fh  = tid >> 12;                               // PIX == 2^12
  const int h   = fh & (HEADS - 1);
  const int frm = fh >> 1;
  const int y = pix >> 6, x = pix & 63;

  const int t = frm * PIX + pix;
  const float scale = 0.17677669529663687f;                // 32^-0.5

  float q[DH];
#pragma unroll
  for (int d = 0; d < DH; ++d)
    q[d] = qkv[(size_t)t * C3 + h * DH + d] * scale;

  int sh = y - KW / 2; sh = sh < 0 ? 0 : (sh > HH - KW ? HH - KW : sh);
  int sw = x - KW / 2; sw = sw < 0 ? 0 : (sw > WW - KW ? WW - KW : sw);

  const float* rb = rpb + h * (2*KW-1) * (2*KW-1)
                        + (sh - y + KW - 1) * (2*KW-1)
                        + (sw - x + KW - 1);

  float m = -3.0e38f, l = 0.f;
  float acc[DH];
#pragma unroll
  for (int d = 0; d < DH; ++d) acc[d] = 0.f;

  for (int a = 0; a < KW; ++a) {
    for (int b = 0; b < KW; ++b) {
      const int kt = frm * PIX + (sh + a) * WW + (sw + b);
      const float* kp = qkv + (size_t)kt * C3 + CC + h * DH;      // K block
      float dot = 0.f;
#pragma unroll
      for (int d = 0; d < DH; ++d) dot = fmaf(q[d], kp[d], dot);
      dot += rb[a * (2*KW-1) + b];

      const float mn   = fmaxf(m, dot);
      const float corr = __expf(m - mn);
      const float p    = __expf(dot - mn);
      l = l * corr + p;
      const float* vp = qkv + (size_t)kt * C3 + 2 * CC + h * DH;  // V block
#pragma unroll
      for (int d = 0; d < DH; ++d) acc[d] = fmaf(acc[d], corr, p * vp[d]);
      m = mn;
    }
  }

  const float inv = 1.f / l;
#pragma unroll
  for (int d = 0; d < DH; ++d)
    attn_t[(size_t)(h * DH + d) * TOK + t] = acc[d] * inv;        // coalesced over pix
}

// ---------------------------------------------------------------------------
// Kernel 3: output projection  (TOK x 64) @ (64 x 64) + b_proj -> d_out
// Same 64x16-per-wave WMMA blocking as kernel 1; attn_t is channel-major so
// A loads coalesce. Also writes the trailing scalar 'inds' output (= 0).
// ---------------------------------------------------------------------------
__global__ __launch_bounds__(256) void proj_gemm_kernel(
    const float* __restrict__ attn_t, const float* __restrict__ w_proj,
    const float* __restrict__ b_proj, float* __restrict__ out)
{
  const int wave = (blockIdx.x * blockDim.x + threadIdx.x) >> 5;
  const int lane = threadIdx.x & 31;
  const int half = lane >> 4;
  const int lr   = lane & 15;

  const int row_blk  = wave / (CC / 16);
  const int col_tile = wave % (CC / 16);
  const int row0 = row_blk * 64;
  const int col0 = col_tile * 16;

  __builtin_prefetch(&w_proj[col0], 0, 0);

  v2f bf[16];
#pragma unroll
  for (int s = 0; s < 16; ++s) {
    const int k0 = 4 * s + 2 * half;
    bf[s].x = w_proj[(k0    ) * CC + col0 + lr];
    bf[s].y = w_proj[(k0 + 1) * CC + col0 + lr];
  }

  const float* aB0 = attn_t + row0 +  0 + lr;   // + c*TOK walks channels
  const float* aB1 = attn_t + row0 + 16 + lr;
  const float* aB2 = attn_t + row0 + 32 + lr;
  const float* aB3 = attn_t + row0 + 48 + lr;

  v8f acc0 = {}, acc1 = {}, acc2 = {}, acc3 = {};
#pragma unroll
  for (int s = 0; s < 16; ++s) {
    const int k0 = 4 * s + 2 * half;
    v2f a0, a1, a2, a3;
    a0.x = aB0[(size_t)(k0) * TOK]; a0.y = aB0[(size_t)(k0 + 1) * TOK];
    a1.x = aB1[(size_t)(k0) * TOK]; a1.y = aB1[(size_t)(k0 + 1) * TOK];
    a2.x = aB2[(size_t)(k0) * TOK]; a2.y = aB2[(size_t)(k0 + 1) * TOK];
    a3.x = aB3[(size_t)(k0) * TOK]; a3.y = aB3[(size_t)(k0 + 1) * TOK];
    acc0 = WMMA_F32(a0, bf[s], acc0);
    acc1 = WMMA_F32(a1, bf[s], acc1);
    acc2 = WMMA_F32(a2, bf[s], acc2);
    acc3 = WMMA_F32(a3, bf[s], acc3);
  }

  const float bias = b_proj[col0 + lr];
#pragma unroll
  for (int r = 0; r < 8; ++r) {
    const int m = r + 8 * half;
    out[(size_t)(row0 +  0 + m) * CC + col0 + lr] = acc0[r] + bias;
    out[(size_t)(row0 + 16 + m) * CC + col0 + lr] = acc1[r] + bias;
    out[(size_t)(row0 + 32 + m) * CC + col0 + lr] = acc2[r] + bias;
    out[(size_t)(row0 + 48 + m) * CC + col0 + lr] = acc3[r] + bias;
  }

  if (blockIdx.x == 0 && threadIdx.x == 0)
    out[(size_t)TOK * CC] = 0.f;    // 'inds' tuple element
}

// ---------------------------------------------------------------------------
extern "C" void kernel_launch(void* const* d_in, const int* in_sizes, int n_in,
                              void* d_out, int out_size, void* d_ws, size_t ws_size,
                              hipStream_t stream)
{
  const float* vid    = (const float*)d_in[0];
  // d_in[1..3] (vid1, fflow, bflow) are unused by the reference computation.
  const float* w_qkv  = (const float*)d_in[4];
  const float* b_qkv  = (const float*)d_in[5];
  const float* rpb    = (const float*)d_in[6];
  const float* w_proj = (const float*)d_in[7];
  const float* b_proj = (const float*)d_in[8];
  float* out = (float*)d_out;

  float* qkv    = (float*)d_ws;                 // TOK*C3 floats  (12.6 MB)
  float* attn_t = qkv + (size_t)TOK * C3;       // CC*TOK floats  ( 4.0 MB)

  // 256 row-blocks (64 rows) x 12 col-tiles = 3072 waves, 8 waves/block
  qkv_gemm_kernel<<<(TOK / 64) * (C3 / 16) / 8, 256, 0, stream>>>(vid, w_qkv, b_qkv, qkv);
  // one thread per (frame, head, pixel) = 32768 threads
  natten_kernel<<<(TOK * HEADS) / 256, 256, 0, stream>>>(qkv, rpb, attn_t);
  // 256 row-blocks x 4 col-tiles = 1024 waves, 8 waves/block
  proj_gemm_kernel<<<(TOK / 64) * (CC / 16) / 8, 256, 0, stream>>>(attn_t, w_proj, b_proj, out);
}